// TropConv2D_70489003262695
// MI455X (gfx1250) — compile-verified
//
#include <hip/hip_runtime.h>
#include <math.h>

// Tropical conv: out[b,ho,wo,f] = max_k(p_k + w_kf) - min_k(p_k + w_kf) + bias_f
// B=8 H=W=32 C=32, 3x3 window stride 1 -> Ho=Wo=30, K=288, F=64.

#define B_   8
#define H_   32
#define W_   32
#define C_   32
#define F_   64
#define KH_  3
#define KW_  3
#define HO_  30
#define WO_  30
#define KTOT (KH_ * KW_ * C_)      // 288
#define NPIX (B_ * HO_ * WO_)      // 7200
#define THREADS 256
#define WAVES   8                  // wave32
#define PIX_PER_WAVE 2
#define PIX_PER_BLK (WAVES * PIX_PER_WAVE)   // 16
#define NBLK (NPIX / PIX_PER_BLK)            // 450 exactly

__global__ __launch_bounds__(THREADS) void trop_conv_kernel(
    const float* __restrict__ x, const float* __restrict__ w,
    const float* __restrict__ bias, float* __restrict__ out)
{
    // Weights resident in LDS: layout identical to global w: [k][f], f fastest.
    __shared__ float wlds[KTOT * F_];   // 72 KB (<= 320 KB WGP LDS)

    const int tid = threadIdx.x;

    // ---- Stage w -> LDS with CDNA5 async global->LDS copies (ASYNCcnt) ----
    {
        unsigned lds_base = (unsigned)(size_t)(&wlds[0]);
        const float* wp = w;
        // 72 KB / (256 threads * 16 B) = 18 b128 transfers per thread
        #pragma unroll
        for (int i = 0; i < (KTOT * F_ * 4) / (THREADS * 16); ++i) {
            unsigned byteoff = ((unsigned)tid + (unsigned)i * THREADS) * 16u;
            unsigned lptr    = lds_base + byteoff;
            // GVS mode: mem_addr = SADDR(64b SGPR) + VADDR(32b byte offset)
            asm volatile("global_load_async_to_lds_b128 %0, %1, %2 offset:0"
                         :: "v"(lptr), "v"(byteoff), "s"(wp)
                         : "memory");
        }
        asm volatile("s_wait_asynccnt 0x0" ::: "memory");
    }
    __syncthreads();

    const int lane = tid & 31;
    // Force wave id into an SGPR so all pixel addressing is provably uniform
    // -> x patch loads become scalar (SMEM) broadcasts.
    const int wave = __builtin_amdgcn_readfirstlane(tid >> 5);

    const float2* __restrict__ wlds2 = (const float2*)wlds;
    const float2  bias2 = ((const float2*)bias)[lane];   // lane owns f = 2*lane, 2*lane+1

    // Each wave handles two pixels concurrently (doubles VALU per DS read).
    const int p0 = blockIdx.x * PIX_PER_BLK + wave * PIX_PER_WAVE;
    const int p1 = p0 + 1;

    const int b0 = p0 / (HO_ * WO_); int r0i = p0 % (HO_ * WO_);
    const int ho0 = r0i / WO_, wo0 = r0i % WO_;
    const int b1 = p1 / (HO_ * WO_); int r1i = p1 % (HO_ * WO_);
    const int ho1 = r1i / WO_, wo1 = r1i % WO_;

    const float* __restrict__ x0 = x + (((b0 * H_) + ho0) * W_ + wo0) * C_;
    const float* __restrict__ x1 = x + (((b1 * H_) + ho1) * W_ + wo1) * C_;

    float mx00 = -INFINITY, mx01 = -INFINITY, mn00 = INFINITY, mn01 = INFINITY;
    float mx10 = -INFINITY, mx11 = -INFINITY, mn10 = INFINITY, mn11 = INFINITY;

    for (int t = 0; t < KH_; ++t) {
        for (int s = 0; s < KW_; ++s) {
            const float* __restrict__ r0 = x0 + (t * W_ + s) * C_;   // 32 contiguous ch, uniform
            const float* __restrict__ r1 = x1 + (t * W_ + s) * C_;
            const float2* __restrict__ wrow = wlds2 + ((t * KW_ + s) * C_) * 32 + lane;
            #pragma unroll
            for (int c = 0; c < C_; ++c) {
                const float  p0v = r0[c];          // wave-uniform -> s_load broadcast
                const float  p1v = r1[c];
                const float2 wv  = wrow[c * 32];   // ds_load_b64, conflict-free
                const float a0 = p0v + wv.x, a1 = p0v + wv.y;
                const float c0 = p1v + wv.x, c1 = p1v + wv.y;
                mx00 = fmaxf(mx00, a0); mn00 = fminf(mn00, a0);
                mx01 = fmaxf(mx01, a1); mn01 = fminf(mn01, a1);
                mx10 = fmaxf(mx10, c0); mn10 = fminf(mn10, c0);
                mx11 = fmaxf(mx11, c1); mn11 = fminf(mn11, c1);
            }
        }
    }

    const float2 o0 = make_float2(mx00 - mn00 + bias2.x, mx01 - mn01 + bias2.y);
    const float2 o1 = make_float2(mx10 - mn10 + bias2.x, mx11 - mn11 + bias2.y);
    ((float2*)out)[p0 * (F_ / 2) + lane] = o0;   // 256 B coalesced store per wave
    ((float2*)out)[p1 * (F_ / 2) + lane] = o1;
}

extern "C" void kernel_launch(void* const* d_in, const int* in_sizes, int n_in,
                              void* d_out, int out_size, void* d_ws, size_t ws_size,
                              hipStream_t stream) {
    const float* x    = (const float*)d_in[0];   // (8,32,32,32) f32
    const float* w    = (const float*)d_in[1];   // (1,1,1,288,64) f32
    const float* bias = (const float*)d_in[2];   // (64,) f32
    float* out = (float*)d_out;                  // (8,30,30,64) f32
    trop_conv_kernel<<<NBLK, THREADS, 0, stream>>>(x, w, bias, out);
}